// SparseLayer_16801912062196
// MI455X (gfx1250) — compile-verified
//
#include <hip/hip_runtime.h>
#include <cstdint>

#define NR_NETS    2048
#define INTERFACE  128
#define WIDTH      128
#define OUT_DIM    64
#define BATCH      256
#define COLS       128        // batch columns per workgroup (2 WGs per net)

// LDS padding vs 64 banks * 4B: ACT_STRIDE%64==8, W_STRIDE%64==4 make the
// WMMA fragment reads conflict-free across the wave32.
#define ACT_STRIDE 136   // 128 + 8 floats
#define W_STRIDE   132   // 128 + 4 floats

typedef float    v2f   __attribute__((ext_vector_type(2)));
typedef float    v8f   __attribute__((ext_vector_type(8)));
typedef uint32_t u32x4 __attribute__((ext_vector_type(4)));
typedef uint32_t u32x8 __attribute__((ext_vector_type(8)));

// ---------------------------------------------------------------------------
// Tensor Data Mover: DMA a 2D fp32 tile (tile_d0 x d1 elements, global row
// stride stride0 elements) into LDS, with D# pad fields inserting the
// bank-conflict-avoiding row padding during the DMA. 2D tensor -> descriptor
// groups 0,1 only (VADDR2/VADDR3 NULL). Inline asm form is portable across
// ROCm7.2 clang-22 and upstream clang-23.
// pad_code_i: pad after 2^(code+1) dwords; pad_code_a: pad amount code+1 dwords
// ---------------------------------------------------------------------------
__device__ __forceinline__ void tdm_load_2d(uint32_t lds_byte_addr,
                                            const float* gptr,
                                            uint32_t tensor_d0, uint32_t tile_d0,
                                            uint32_t d1, uint32_t stride0,
                                            uint32_t pad_code_i,
                                            uint32_t pad_code_a) {
  uint64_t ga = (uint64_t)(uintptr_t)gptr;
  u32x4 g0;
  g0.s0 = 1u;                                                // count=1 valid, user desc
  g0.s1 = lds_byte_addr;                                     // lds_addr
  g0.s2 = (uint32_t)ga;                                      // global_addr[31:0]
  g0.s3 = ((uint32_t)(ga >> 32) & 0x01FFFFFFu) | (2u << 30); // addr[56:32] | type=2
  u32x8 g1;
  g1.s0 = (2u << 16)                 // data_size = 4B
        | (1u << 20)                 // pad_enable
        | (pad_code_i << 22)         // pad_interval
        | (pad_code_a << 25);        // pad_amount
  g1.s1 = (tensor_d0 & 0xFFFFu) << 16;                    // tensor_dim0[15:0]
  g1.s2 = (tensor_d0 >> 16) | ((d1 & 0xFFFFu) << 16);     // dim0[31:16] | dim1[15:0]
  g1.s3 = (d1 >> 16) | ((tile_d0 & 0xFFFFu) << 16);       // dim1[31:16] | tile_dim0
  g1.s4 = (d1 & 0xFFFFu);                                 // tile_dim1 (tile_dim2=0)
  g1.s5 = stride0;                                        // tensor_dim0_stride (elems)
  g1.s6 = 0u;
  g1.s7 = 0u;
  asm volatile("tensor_load_to_lds %0, %1" :: "s"(g0), "s"(g1) : "memory");
}

// ---------------------------------------------------------------------------
// 2(mtile) x NT(ntile) block of 16x16 output tiles via V_WMMA_F32_16X16X4_F32:
//   acc += W[32 rows, 128] * Act[128, NT*16 cols]
// A frag (16x4 f32): lane<16 -> {K,K+1}, lane>=16 -> {K+2,K+3}, row = lane&15
// B frag (4x16 f32): same K split, col = lane&15
// ---------------------------------------------------------------------------
template <int NT>
__device__ __forceinline__ void gemm_tiles(const float* __restrict__ wlds,
                                           const float* __restrict__ act,
                                           int mbase, int n0, int l16, int half,
                                           v8f acc[2][NT]) {
#pragma unroll
  for (int mi = 0; mi < 2; ++mi)
#pragma unroll
    for (int ni = 0; ni < NT; ++ni)
      acc[mi][ni] = v8f{0.f, 0.f, 0.f, 0.f, 0.f, 0.f, 0.f, 0.f};

  const float* wr0 = wlds + (mbase + l16) * W_STRIDE;
  const float* wr1 = wlds + (mbase + 16 + l16) * W_STRIDE;
  const float* bcol = act + n0 + l16;

#pragma unroll 2
  for (int k = 0; k < WIDTH; k += 4) {
    const int kk = k + 2 * half;
    v2f a0 = *(const v2f*)(wr0 + kk);           // ds_load_b64, all 64 banks once
    v2f a1 = *(const v2f*)(wr1 + kk);
    const float* b0 = bcol + kk * ACT_STRIDE;
#pragma unroll
    for (int ni = 0; ni < NT; ++ni) {
      v2f b;
      b.x = b0[ni * 16];                        // row kk
      b.y = b0[ni * 16 + ACT_STRIDE];           // row kk+1
      acc[0][ni] = __builtin_amdgcn_wmma_f32_16x16x4_f32(
          false, a0, false, b, (short)0, acc[0][ni], false, false);
      acc[1][ni] = __builtin_amdgcn_wmma_f32_16x16x4_f32(
          false, a1, false, b, (short)0, acc[1][ni], false, false);
    }
  }
}

template <int NT>
__device__ __forceinline__ void store_tiles_lds(float* __restrict__ act,
                                                int mbase, int n0, int l16,
                                                int half, v8f acc[2][NT]) {
#pragma unroll
  for (int mi = 0; mi < 2; ++mi)
#pragma unroll
    for (int ni = 0; ni < NT; ++ni)
#pragma unroll
      for (int j = 0; j < 8; ++j)
        act[(mbase + mi * 16 + 8 * half + j) * ACT_STRIDE + n0 + ni * 16 + l16] =
            acc[mi][ni][j];
}

// ---------------------------------------------------------------------------
// One workgroup = one net x one 128-column batch half. 8 waves.
// LDS: 128*136*4 + 128*132*4 = 69632 + 67584 = 137216 B -> 2 WGs per 320KB WGP,
// so one WG's TDM waits overlap the co-resident WG's WMMA compute.
// All three layers fused; activation stays in LDS; W0->W1->W2 stream through a
// single weight buffer (the sibling WG hides each DMA behind its compute, and
// the sibling's weight re-fetch hits the 192MB L2).
// ---------------------------------------------------------------------------
__global__ __launch_bounds__(256, 4) void fused_block_mlp_kernel(
    const float* __restrict__ x,  const float* __restrict__ w0,
    const float* __restrict__ w1, const float* __restrict__ w2,
    float* __restrict__ y) {
  __shared__ __align__(16) float s_act[INTERFACE * ACT_STRIDE];
  __shared__ __align__(16) float s_w[WIDTH * W_STRIDE];

  const int bid  = blockIdx.x;
  const int net  = bid >> 1;
  const int c0   = (bid & 1) * COLS;   // batch-column half
  const int tid  = threadIdx.x;
  const int lane = tid & 31;
  const int l16  = lane & 15;
  const int half = lane >> 4;
  const int wv   = tid >> 5;

  const uint32_t act_lds = (uint32_t)(uintptr_t)&s_act[0];
  const uint32_t w_lds   = (uint32_t)(uintptr_t)&s_w[0];

  // X half-tile (strided rows) + W0, issued by wave 0 (TDM ignores EXEC).
  if (wv == 0) {
    tdm_load_2d(act_lds, x + (size_t)net * INTERFACE * BATCH + c0,
                BATCH, COLS, INTERFACE, BATCH, 6, 7);              // -> stride 136
    tdm_load_2d(w_lds, w0 + (size_t)net * WIDTH * INTERFACE,
                INTERFACE, INTERFACE, WIDTH, INTERFACE, 6, 3);     // -> stride 132
    __builtin_amdgcn_s_wait_tensorcnt(0);
  }
  __syncthreads();

  // Layers 0/1: 8x8 tiles of 16x16 -> 2m x 4n per wave.
  const int mb01 = 32 * (wv >> 1);
  const int nb01 = 64 * (wv & 1);
  v8f acc[2][4];

  // ---- Layer 0: H0 = W0 * X
  gemm_tiles<4>(s_w, s_act, mb01, nb01, l16, half, acc);
  __syncthreads();                                   // all reads of W0/act done
  if (wv == 0)
    tdm_load_2d(w_lds, w1 + (size_t)net * WIDTH * WIDTH,
                WIDTH, WIDTH, WIDTH, WIDTH, 6, 3);   // W1 DMA behind writeback
  store_tiles_lds<4>(s_act, mb01, nb01, l16, half, acc);
  if (wv == 0) __builtin_amdgcn_s_wait_tensorcnt(0);
  __syncthreads();

  // ---- Layer 1: H1 = W1 * H0
  gemm_tiles<4>(s_w, s_act, mb01, nb01, l16, half, acc);
  __syncthreads();
  if (wv == 0)
    tdm_load_2d(w_lds, w2 + (size_t)net * OUT_DIM * WIDTH,
                WIDTH, WIDTH, OUT_DIM, WIDTH, 6, 3); // W2 DMA behind writeback
  store_tiles_lds<4>(s_act, mb01, nb01, l16, half, acc);
  if (wv == 0) __builtin_amdgcn_s_wait_tensorcnt(0);
  __syncthreads();

  // ---- Layer 2: Y = W2 * H1 -> 4x8 tiles -> 2m x 2n per wave, straight to HBM
  const int mb2 = 32 * (wv >> 2);
  const int nb2 = 32 * (wv & 3);
  v8f acc2[2][2];
  gemm_tiles<2>(s_w, s_act, mb2, nb2, l16, half, acc2);

  float* yb = y + (size_t)net * OUT_DIM * BATCH + c0;
#pragma unroll
  for (int mi = 0; mi < 2; ++mi)
#pragma unroll
    for (int ni = 0; ni < 2; ++ni)
#pragma unroll
      for (int j = 0; j < 8; ++j)
        yb[(mb2 + mi * 16 + 8 * half + j) * BATCH + nb2 + ni * 16 + l16] =
            acc2[mi][ni][j];
}

extern "C" void kernel_launch(void* const* d_in, const int* in_sizes, int n_in,
                              void* d_out, int out_size, void* d_ws, size_t ws_size,
                              hipStream_t stream) {
  const float* x  = (const float*)d_in[0];
  const float* w0 = (const float*)d_in[1];
  const float* w1 = (const float*)d_in[2];
  const float* w2 = (const float*)d_in[3];
  float* y = (float*)d_out;
  fused_block_mlp_kernel<<<NR_NETS * 2, 256, 0, stream>>>(x, w0, w1, w2, y);
}